// DiffusionSystem_47150150976005
// MI455X (gfx1250) — compile-verified
//
#include <hip/hip_runtime.h>
#include <hip/hip_bf16.h>
#include <stdint.h>

// Problem constants from the reference: state is [4, 8, 1024, 1024] f32.
#define HH 1024
#define WW 1024
#define RROWS 8                   // interior rows computed per block
#define TILE_ROWS (RROWS + 2)     // + top/bottom halo
#define NTHREADS 256              // 8 waves (wave32)

typedef __attribute__((ext_vector_type(4))) unsigned int tdm_g0_t;
typedef __attribute__((ext_vector_type(8))) unsigned int tdm_g1_t;

// ---------------------------------------------------------------------------
// One TDM descriptor: DMA `nrows` contiguous rows of 1024 f32 from global
// memory into LDS at byte offset `lds_off`. D# layout per CDNA5 ISA §8:
//   group0: [1:0]=count=1, [63:32]=lds_addr, [120:64]=global_addr, [127:126]=2
//   group1: [17:16]=data_size(4B), tensor_dim0=1024, tensor_dim1=nrows,
//           tile_dim0=1024, tile_dim1=nrows, tensor_dim0_stride=1024
// Two-group form => VADDR2/VADDR3 NULL => 2D tensor. Tracked by TENSORcnt.
// ---------------------------------------------------------------------------
__device__ __forceinline__ void tdm_load_rows(const float* gsrc,
                                              unsigned int lds_off,
                                              unsigned int nrows) {
  const unsigned long long ga = (unsigned long long)(uintptr_t)gsrc;
  const tdm_g0_t g0 = {
      1u,                                                   // count=1 (user D#)
      lds_off,                                              // lds_addr (bytes)
      (unsigned int)ga,                                     // global_addr[31:0]
      (unsigned int)((ga >> 32) & 0x1FFFFFFu) | 0x80000000u // addr[56:32]|type=2
  };
  const tdm_g1_t g1 = {
      0x20000u,                 // wg_mask=0, data_size=2 (4B), no pad/iterate
      (WW & 0xFFFFu) << 16,     // tensor_dim0 lo16 (1024)
      nrows << 16,              // tensor_dim0 hi16 (0) | tensor_dim1 lo16
      (WW & 0xFFFFu) << 16,     // tensor_dim1 hi16 (0) | tile_dim0 = 1024
      nrows,                    // tile_dim1 = nrows, tile_dim2 = 0
      (unsigned int)WW,         // tensor_dim0_stride lo32 = 1024 elements
      0u,                       // dim0_stride hi16 | dim1_stride lo16
      0u                        // dim1_stride hi32
  };
  asm volatile("tensor_load_to_lds %0, %1" :: "s"(g0), "s"(g1) : "memory");
}

// Stage TILE_ROWS rows (periodic y-wrap) of one [H,W] slice into LDS via the
// Tensor Data Mover. Issued by wave 0 only (TDM ignores EXEC; TENSORcnt is
// per-wave), then published to the workgroup with a barrier.
__device__ __forceinline__ void load_tile_tdm(float (*tile)[WW],
                                              const float* __restrict__ base,
                                              int r0) {
  if (threadIdx.x < 32) {                       // wave-uniform branch: wave 0
    const int rowStart = r0 - 1;
    if (rowStart < 0) {
      // top edge: wrapped row H-1 -> tile[0], rows 0..RROWS -> tile[1..]
      tdm_load_rows(base + (size_t)(HH - 1) * WW,
                    (unsigned int)(uintptr_t)&tile[0][0], 1u);
      tdm_load_rows(base, (unsigned int)(uintptr_t)&tile[1][0],
                    (unsigned int)(TILE_ROWS - 1));
    } else if (r0 + RROWS >= HH) {
      // bottom edge: rows r0-1..H-1 -> tile[0..R], wrapped row 0 -> tile[R+1]
      tdm_load_rows(base + (size_t)rowStart * WW,
                    (unsigned int)(uintptr_t)&tile[0][0],
                    (unsigned int)(TILE_ROWS - 1));
      tdm_load_rows(base, (unsigned int)(uintptr_t)&tile[TILE_ROWS - 1][0], 1u);
    } else {
      // interior: one 10-row descriptor moves the whole 40 KB tile
      tdm_load_rows(base + (size_t)rowStart * WW,
                    (unsigned int)(uintptr_t)&tile[0][0],
                    (unsigned int)TILE_ROWS);
    }
    __builtin_amdgcn_s_wait_tensorcnt(0);       // s_wait_tensorcnt 0x0
  }
  __syncthreads();                              // LDS visible to all 8 waves
}

// wave32 shuffle reduction of a double
__device__ __forceinline__ double wave_reduce(double v) {
#pragma unroll
  for (int o = 16; o > 0; o >>= 1) v += __shfl_down(v, o, 32);
  return v;
}

__global__ void zero_ws_kernel(double* ws, int n) {
  int i = blockIdx.x * blockDim.x + threadIdx.x;
  if (i < n) ws[i] = 0.0;
}

// ---------------------------------------------------------------------------
// Pass A: f64 stencil; accumulate per-slice initial mass (raw input) and
// final mass (diffused) into ws[slice] / ws[nslices+slice].
// ---------------------------------------------------------------------------
__global__ __launch_bounds__(NTHREADS)
void diffuse_mass_kernel(const float* __restrict__ s,
                         const double* __restrict__ dcoef,
                         const double* __restrict__ dtp,
                         double* __restrict__ mass, int nslices) {
  __shared__ float tile[TILE_ROWS][WW];
  __shared__ double red[2 * (NTHREADS / 32)];

  const int blocksPerSlice = HH / RROWS;
  const int slice = blockIdx.x / blocksPerSlice;
  const int r0 = (blockIdx.x % blocksPerSlice) * RROWS;
  const float* base = s + (size_t)slice * HH * WW;

  load_tile_tdm(tile, base, r0);

  const double scale = dcoef[0] * dtp[0];
  const double wE = scale * (1.0 / 6.0);     // edge weights
  const double wK = scale * (1.0 / 12.0);    // corner weights
  const double wM = 1.0 - scale;             // center: kernel sums to 1

  const int x0 = threadIdx.x * 4;
  double sum_in = 0.0, sum_out = 0.0;
  for (int yy = 1; yy <= RROWS; ++yy) {
#pragma unroll
    for (int j = 0; j < 4; ++j) {
      const int x = x0 + j;
      const int xm = (x - 1) & (WW - 1);     // periodic wrap in x (free in LDS)
      const int xp = (x + 1) & (WW - 1);
      const double c  = (double)tile[yy][x];
      const double nn = (double)tile[yy - 1][x];
      const double ss = (double)tile[yy + 1][x];
      const double ee = (double)tile[yy][xp];
      const double wwv = (double)tile[yy][xm];
      const double ne = (double)tile[yy - 1][xp];
      const double nw = (double)tile[yy - 1][xm];
      const double se = (double)tile[yy + 1][xp];
      const double sw = (double)tile[yy + 1][xm];
      const double d = wM * c + wE * (nn + ss + ee + wwv) + wK * (ne + nw + se + sw);
      sum_in += c;
      sum_out += d;
    }
  }

  // block reduction: wave shuffle -> LDS partials -> one atomic per quantity
  sum_in = wave_reduce(sum_in);
  sum_out = wave_reduce(sum_out);
  const int lane = threadIdx.x & 31;
  const int wv = threadIdx.x >> 5;
  if (lane == 0) {
    red[wv] = sum_in;
    red[(NTHREADS / 32) + wv] = sum_out;
  }
  __syncthreads();
  if (threadIdx.x == 0) {
    double ti = 0.0, to = 0.0;
#pragma unroll
    for (int k = 0; k < NTHREADS / 32; ++k) {
      ti += red[k];
      to += red[(NTHREADS / 32) + k];
    }
    atomicAdd(&mass[slice], ti);
    atomicAdd(&mass[nslices + slice], to);
  }
}

// ---------------------------------------------------------------------------
// Pass B: recompute the stencil (input fits in the 192 MB L2, so re-reads are
// L2 hits), add the per-slice mass correction, store f32 as b128.
// ---------------------------------------------------------------------------
__global__ __launch_bounds__(NTHREADS)
void diffuse_write_kernel(const float* __restrict__ s,
                          const double* __restrict__ dcoef,
                          const double* __restrict__ dtp,
                          const double* __restrict__ mass,
                          float* __restrict__ out, int nslices) {
  __shared__ float tile[TILE_ROWS][WW];

  const int blocksPerSlice = HH / RROWS;
  const int slice = blockIdx.x / blocksPerSlice;
  const int r0 = (blockIdx.x % blocksPerSlice) * RROWS;
  const float* base = s + (size_t)slice * HH * WW;

  load_tile_tdm(tile, base, r0);

  const double scale = dcoef[0] * dtp[0];
  const double wE = scale * (1.0 / 6.0);
  const double wK = scale * (1.0 / 12.0);
  const double wM = 1.0 - scale;
  const double corr = (mass[slice] - mass[nslices + slice]) *
                      (1.0 / ((double)HH * (double)WW));

  const int x0 = threadIdx.x * 4;
  float* obase = out + (size_t)slice * HH * WW;
  for (int yy = 1; yy <= RROWS; ++yy) {
    float4 o;
    float r[4];
#pragma unroll
    for (int j = 0; j < 4; ++j) {
      const int x = x0 + j;
      const int xm = (x - 1) & (WW - 1);
      const int xp = (x + 1) & (WW - 1);
      const double c  = (double)tile[yy][x];
      const double nn = (double)tile[yy - 1][x];
      const double ss = (double)tile[yy + 1][x];
      const double ee = (double)tile[yy][xp];
      const double wwv = (double)tile[yy][xm];
      const double ne = (double)tile[yy - 1][xp];
      const double nw = (double)tile[yy - 1][xm];
      const double se = (double)tile[yy + 1][xp];
      const double sw = (double)tile[yy + 1][xm];
      const double d = wM * c + wE * (nn + ss + ee + wwv) + wK * (ne + nw + se + sw);
      r[j] = (float)(d + corr);
    }
    o.x = r[0]; o.y = r[1]; o.z = r[2]; o.w = r[3];
    *(float4*)(obase + (size_t)(r0 + yy - 1) * WW + x0) = o;  // b128 store
  }
}

extern "C" void kernel_launch(void* const* d_in, const int* in_sizes, int n_in,
                              void* d_out, int out_size, void* d_ws, size_t ws_size,
                              hipStream_t stream) {
  const float* s = (const float*)d_in[0];
  const double* dc = (const double*)d_in[1];   // float64 scalar
  const double* dt = (const double*)d_in[2];   // float64 scalar
  float* out = (float*)d_out;

  const int n = in_sizes[0];                   // 32 * 1024 * 1024
  const int nslices = n / (HH * WW);           // 32 (B*C)
  double* mass = (double*)d_ws;                // 2 * nslices doubles

  zero_ws_kernel<<<1, 64, 0, stream>>>(mass, 2 * nslices);

  const int nblocks = nslices * (HH / RROWS);
  diffuse_mass_kernel<<<nblocks, NTHREADS, 0, stream>>>(s, dc, dt, mass, nslices);
  diffuse_write_kernel<<<nblocks, NTHREADS, 0, stream>>>(s, dc, dt, mass, out, nslices);
}